// InstanceContrastiveLoss_56573309223630
// MI455X (gfx1250) — compile-verified
//
#include <hip/hip_runtime.h>
#include <math.h>

#define NROWS 8192
#define NDIM 512
#define INV_T (1.0f / 0.07f)
#define ASTRIDE 514   // 512 + 2 pad: stride mod 64 banks = 2 -> conflict-free v2f LDS reads
#define RPB 32        // rows per block: 2 row-tiles of 16 per wave

typedef float v2f __attribute__((ext_vector_type(2)));
typedef float v8f __attribute__((ext_vector_type(8)));

// ---------------------------------------------------------------------------
// Pass 0: L2-normalize rows. One wave (32 lanes) per row, 16 floats per lane.
// ---------------------------------------------------------------------------
__global__ __launch_bounds__(256) void icl_normalize(const float* __restrict__ feat,
                                                     float* __restrict__ fn) {
  const int wid  = threadIdx.x >> 5;
  const int lane = threadIdx.x & 31;
  const int row  = blockIdx.x * 8 + wid;
  const float4* src = (const float4*)(feat + (size_t)row * NDIM);
  float4*       dst = (float4*)(fn + (size_t)row * NDIM);
  float4 v[4];
  float ss = 0.f;
#pragma unroll
  for (int q = 0; q < 4; ++q) {
    v[q] = src[q * 32 + lane];
    ss += v[q].x * v[q].x + v[q].y * v[q].y + v[q].z * v[q].z + v[q].w * v[q].w;
  }
#pragma unroll
  for (int off = 16; off >= 1; off >>= 1) ss += __shfl_xor(ss, off, 32);
  const float sc = 1.0f / fmaxf(sqrtf(ss), 1e-12f);
#pragma unroll
  for (int q = 0; q < 4; ++q) {
    v[q].x *= sc; v[q].y *= sc; v[q].z *= sc; v[q].w *= sc;
    dst[q * 32 + lane] = v[q];
  }
}

// ---------------------------------------------------------------------------
// Pass 1: fused sim-GEMM (fp32 WMMA 16x16x4, 2-way row register blocking)
// + per-row top5 / positive stats. Block = 512 threads = 16 waves, owns 32
// rows; each wave computes 2 row-tiles against 32 column tiles, so every B
// fragment load feeds two WMMAs.
// ---------------------------------------------------------------------------
__global__ __launch_bounds__(512) void icl_sim_stats(const float* __restrict__ fn,
                                                     const int* __restrict__ idx,
                                                     float* __restrict__ rs) {
  __shared__ float As[RPB * ASTRIDE];     // A strip, padded rows (~64KB)
  __shared__ float Cb[16][16][17];        // per-wave 16x16 tile spill (reused)
  __shared__ float Mb[16][RPB][10];       // per-wave partial row stats

  const int tid  = threadIdx.x;
  const int wid  = tid >> 5;
  const int lane = tid & 31;
  const int h    = lane >> 4;             // half-wave select (K pairs)
  const int l16  = lane & 15;
  const int row0 = blockIdx.x * RPB;

  // Stage 32x512 A strip into LDS (v2f granules keep 8B alignment w/ stride 514)
  {
    const v2f* src = (const v2f*)(fn + (size_t)row0 * NDIM);
    for (int i = tid; i < RPB * NDIM / 2; i += 512) {
      const int r = i >> 8;               // row (256 v2f per row)
      const int c = i & 255;              // v2f col
      *(v2f*)(As + r * ASTRIDE + c * 2) = src[i];
    }
  }
  __syncthreads();

  const int ridx[2] = { idx[row0 + l16], idx[row0 + 16 + l16] };
  const int grow[2] = { row0 + l16, row0 + 16 + l16 };
  float t[2][5], m[2], s[2], sump[2];
  int   cnt[2];
#pragma unroll
  for (int rt = 0; rt < 2; ++rt) {
#pragma unroll
    for (int j = 0; j < 5; ++j) t[rt][j] = -__builtin_inff();
    m[rt] = -__builtin_inff(); s[rt] = 0.f; sump[rt] = 0.f; cnt[rt] = 0;
  }

  for (int ct = wid; ct < NROWS / 16; ct += 16) {
    const int col0  = ct * 16;
    const int cidxv = idx[col0 + l16];

    v8f acc[2] = { {}, {} };
    const float* Ap0 = As + l16 * ASTRIDE + 2 * h;
    const float* Ap1 = Ap0 + 16 * ASTRIDE;
    const float* Bp  = fn + (size_t)(col0 + l16) * NDIM + 2 * h;
#pragma unroll 4
    for (int k = 0; k < NDIM; k += 4) {
      v2f b  = *(const v2f*)(Bp + k);     // shared by both row-tiles
      v2f a0 = *(const v2f*)(Ap0 + k);
      v2f a1 = *(const v2f*)(Ap1 + k);
      acc[0] = __builtin_amdgcn_wmma_f32_16x16x4_f32(false, a0, false, b,
                                                     (short)0, acc[0], false, false);
      acc[1] = __builtin_amdgcn_wmma_f32_16x16x4_f32(false, a1, false, b,
                                                     (short)0, acc[1], false, false);
    }

#pragma unroll
    for (int rt = 0; rt < 2; ++rt) {
      // Spill tile: VGPR v holds row (v + 8h), col l16
#pragma unroll
      for (int v = 0; v < 8; ++v) Cb[wid][v + 8 * h][l16] = acc[rt][v];
      asm volatile("s_wait_dscnt 0" ::: "memory");  // order spill vs cross-lane scan

      if (h == 0) {  // lanes 0..15 each own one row of the tile
#pragma unroll 4
        for (int c = 0; c < 16; ++c) {
          const float val  = Cb[wid][l16][c];
          const int   gcol = col0 + c;
          const int   cidx = __shfl(cidxv, c, 32);
          if (gcol != grow[rt]) {
            // streaming top-5 (off-diagonal, positives included, per reference)
            if (val > t[rt][4]) {
              t[rt][4] = val;
#pragma unroll
              for (int j = 4; j > 0; --j)
                if (t[rt][j] > t[rt][j - 1]) {
                  float x = t[rt][j - 1]; t[rt][j - 1] = t[rt][j]; t[rt][j] = x;
                }
            }
            if (cidx == ridx[rt]) {  // positive: online softmax of sim/T
              cnt[rt]++; sump[rt] += val;
              const float scv = val * INV_T;
              if (scv > m[rt]) { s[rt] *= expf(m[rt] - scv); m[rt] = scv; }
              s[rt] += expf(scv - m[rt]);
            }
          }
        }
      }
      // per-wave LDS ops are in-order: next rt's spill won't pass these loads
    }
  }

  if (h == 0) {
#pragma unroll
    for (int rt = 0; rt < 2; ++rt) {
      float* p = &Mb[wid][l16 + 16 * rt][0];
      p[0] = t[rt][0]; p[1] = t[rt][1]; p[2] = t[rt][2]; p[3] = t[rt][3]; p[4] = t[rt][4];
      p[5] = m[rt];    p[6] = s[rt];    p[7] = (float)cnt[rt]; p[8] = sump[rt];
    }
  }
  __syncthreads();

  if (wid == 0 && h == 0) {  // merge 16 wave-partials; each lane merges 2 rows
    for (int half = 0; half < 2; ++half) {
      const int r = l16 + 16 * half;
      float* p = &Mb[0][r][0];
      float T0=p[0],T1=p[1],T2=p[2],T3=p[3],T4=p[4],M=p[5],S=p[6],CNT=p[7],SP=p[8];
      for (int w = 1; w < 16; ++w) {
        float* q = &Mb[w][r][0];
#pragma unroll
        for (int k = 0; k < 5; ++k) {
          const float val = q[k];
          if (val > T4) {
            T4 = val;
            if (T4 > T3) { float x = T3; T3 = T4; T4 = x; }
            if (T3 > T2) { float x = T2; T2 = T3; T3 = x; }
            if (T2 > T1) { float x = T1; T1 = T2; T2 = x; }
            if (T1 > T0) { float x = T0; T0 = T1; T1 = x; }
          }
        }
        const float m2 = q[5], s2 = q[6];
        if (s2 > 0.f) {
          if (!(S > 0.f)) { M = m2; S = s2; }
          else {
            const float nm = fmaxf(M, m2);
            S = S * expf(M - nm) + s2 * expf(m2 - nm);
            M = nm;
          }
        }
        CNT += q[7]; SP += q[8];
      }
      float* o = rs + (size_t)(row0 + r) * 12;
      o[0]=T0; o[1]=T1; o[2]=T2; o[3]=T3; o[4]=T4;
      o[5]=M;  o[6]=S;  o[7]=CNT; o[8]=SP;
    }
  }
}

// ---------------------------------------------------------------------------
// Pass 2: global max of hard scores, per-row loss, mean. Single block.
// ---------------------------------------------------------------------------
__global__ __launch_bounds__(1024) void icl_finalize(const float* __restrict__ rs,
                                                     float* __restrict__ out) {
  __shared__ float red[32];
  __shared__ float gshare;
  const int tid  = threadIdx.x;
  const int wid  = tid >> 5;
  const int lane = tid & 31;

  float g = -__builtin_inff();
  for (int i = tid; i < NROWS; i += 1024) g = fmaxf(g, rs[(size_t)i * 12]);
#pragma unroll
  for (int off = 16; off >= 1; off >>= 1) g = fmaxf(g, __shfl_xor(g, off, 32));
  if (lane == 0) red[wid] = g;
  __syncthreads();
  if (wid == 0) {
    float v = red[lane];
#pragma unroll
    for (int off = 16; off >= 1; off >>= 1) v = fmaxf(v, __shfl_xor(v, off, 32));
    if (lane == 0) gshare = v;
  }
  __syncthreads();
  const float gsc = gshare * INV_T;  // global max of hard scores (reference semantics)

  float sum = 0.f;
  for (int i = tid; i < NROWS; i += 1024) {
    const float* p = rs + (size_t)i * 12;
    const float hh = expf(p[0] * INV_T - gsc) + expf(p[1] * INV_T - gsc) +
                     expf(p[2] * INV_T - gsc) + expf(p[3] * INV_T - gsc) +
                     expf(p[4] * INV_T - gsc);
    const float S = p[6], CNT = p[7], SP = p[8];
    float per = 0.f;
    if (CNT > 0.5f) per = logf(S + hh) - SP * INV_T / CNT;  // mixed-shift denom, per ref
    sum += per;
  }
#pragma unroll
  for (int off = 16; off >= 1; off >>= 1) sum += __shfl_xor(sum, off, 32);
  if (lane == 0) red[wid] = sum;
  __syncthreads();
  if (wid == 0) {
    float v = red[lane];
#pragma unroll
    for (int off = 16; off >= 1; off >>= 1) v += __shfl_xor(v, off, 32);
    if (lane == 0) out[0] = v / (float)NROWS;
  }
}

// ---------------------------------------------------------------------------
extern "C" void kernel_launch(void* const* d_in, const int* in_sizes, int n_in,
                              void* d_out, int out_size, void* d_ws, size_t ws_size,
                              hipStream_t stream) {
  const float* feat = (const float*)d_in[0];
  const int*   idx  = (const int*)d_in[1];
  float*       out  = (float*)d_out;

  float* Fn = (float*)d_ws;                       // 8192*512 normalized features
  float* RS = Fn + (size_t)NROWS * NDIM;          // 8192*12 per-row stats

  icl_normalize<<<NROWS / 8, 256, 0, stream>>>(feat, Fn);
  icl_sim_stats<<<NROWS / RPB, 512, 0, stream>>>(Fn, idx, RS);
  icl_finalize<<<1, 1024, 0, stream>>>(RS, out);
}